// IC_Adjust_46136538693752
// MI455X (gfx1250) — compile-verified
//
#include <hip/hip_runtime.h>
#include <hip/hip_bf16.h>

// out[i, j] = x[i, j] * a[j] ; x: [N, 1024] fp32, a: [1024] fp32.
// Bandwidth-bound (0.125 FLOP/byte): floor = 537 MB / 23.3 TB/s ~= 23 us.
// CDNA5 features used: async global->LDS staging of the scale vector
// (ASYNCcnt path), B128 accesses with NT hints (both streams > 192 MB L2),
// uniform saddr + immediate-offset addressing for clause-friendly codegen.

typedef __attribute__((ext_vector_type(4))) float v4f;

#define XSIZE          1024
#define VECS_PER_ROW   (XSIZE / 4)   // 256 float4 per row -> one per thread
#define ROWS_PER_BLOCK 8             // 8 B128 NT loads in flight per thread

__global__ __launch_bounds__(VECS_PER_ROW) void
ic_adjust_scale_cols(const float* __restrict__ x, const float* __restrict__ a,
                     float* __restrict__ out, int nrows) {
    __shared__ __align__(16) float a_lds[XSIZE];

    const int tid = threadIdx.x;     // 0..255 : owns columns [4*tid, 4*tid+4)

    // ---- one-time stage of the 4 KB scale vector via the CDNA5 async path ----
    {
        unsigned lds_off = (unsigned)(uintptr_t)(&a_lds[0]) + (unsigned)(tid * 16);
        unsigned long long gaddr =
            (unsigned long long)(uintptr_t)a + (unsigned long long)(tid * 16);
        asm volatile("global_load_async_to_lds_b128 %0, %1, off"
                     :: "v"(lds_off), "v"(gaddr)
                     : "memory");
        asm volatile("s_wait_asynccnt 0" ::: "memory");
    }
    __syncthreads();

    const v4f av = *reinterpret_cast<const v4f*>(&a_lds[tid * 4]);  // ds_load_b128

    const int row0 = (int)blockIdx.x * ROWS_PER_BLOCK;
    if (row0 >= nrows) return;

    // Uniform (scalar) 64-bit base for this block's row tile; per-lane address
    // is then 32-bit tid*16 + compile-time immediate i*4096 -> saddr+voffset.
    const v4f* __restrict__ xb =
        reinterpret_cast<const v4f*>(x) + (long)row0 * VECS_PER_ROW;
    v4f* __restrict__ ob =
        reinterpret_cast<v4f*>(out) + (long)row0 * VECS_PER_ROW;

    if (row0 + ROWS_PER_BLOCK <= nrows) {
        // ---- full tile: 8 clause-able NT B128 loads, 8 muls, 8 NT stores ----
        v4f r[ROWS_PER_BLOCK];
#pragma unroll
        for (int i = 0; i < ROWS_PER_BLOCK; ++i)
            r[i] = __builtin_nontemporal_load(&xb[i * VECS_PER_ROW + tid]);
#pragma unroll
        for (int i = 0; i < ROWS_PER_BLOCK; ++i)
            r[i] *= av;
#pragma unroll
        for (int i = 0; i < ROWS_PER_BLOCK; ++i)
            __builtin_nontemporal_store(r[i], &ob[i * VECS_PER_ROW + tid]);
    } else {
        // ---- tail tile (not hit for N = 65536, kept for generality) ----
        const int rem = nrows - row0;
        for (int i = 0; i < rem; ++i) {
            v4f r = __builtin_nontemporal_load(&xb[i * VECS_PER_ROW + tid]);
            r *= av;
            __builtin_nontemporal_store(r, &ob[i * VECS_PER_ROW + tid]);
        }
    }
}

extern "C" void kernel_launch(void* const* d_in, const int* in_sizes, int n_in,
                              void* d_out, int out_size, void* d_ws, size_t ws_size,
                              hipStream_t stream) {
    const float* x = (const float*)d_in[0];   // [N, 1024] fp32
    const float* a = (const float*)d_in[1];   // [1024] fp32
    float* out = (float*)d_out;

    const int xsize = in_sizes[1];            // 1024
    const int nrows = in_sizes[0] / xsize;    // 65536

    dim3 block(VECS_PER_ROW);                 // 256 threads = 8 wave32
    dim3 grid((nrows + ROWS_PER_BLOCK - 1) / ROWS_PER_BLOCK);  // 8192 blocks

    ic_adjust_scale_cols<<<grid, block, 0, stream>>>(x, a, out, nrows);
}